// RelativeCoordPredictor_52561809769060
// MI455X (gfx1250) — compile-verified
//
#include <hip/hip_runtime.h>
#include <hip/hip_bf16.h>
#include <math.h>

// Problem constants (from reference)
#define NB   64
#define CCH  1024
#define HH   32
#define WW   32
#define HWSZ 1024          // H*W
#define SIZEF 32.0f

// Output offsets (flat concat, floats)
#define OUT0_OFF 0                       // relative_coord_total [N,HW,2]
#define OUT1_OFF (NB*HWSZ*2)             // basic_anchor [N,2]        = 131072
#define OUT2_OFF (OUT1_OFF + NB*2)       // position_weight [N,HW,HW] = 131200
#define OUT3_OFF (OUT2_OFF + (size_t)NB*HWSZ*HWSZ)  // argmax idx [N] = 67240064

typedef __attribute__((ext_vector_type(2))) float v2f;
typedef __attribute__((ext_vector_type(8))) float v8f;

// ---------------------------------------------------------------------------
// Kernel 1: chansum[n,hw] = sum_c x[n,c,hw]   (single pass over x, 268 MB)
// grid: N*HW/256 = 256 blocks x 256 threads; coalesced across hw.
// ---------------------------------------------------------------------------
__global__ void rcp_chansum_kernel(const float* __restrict__ x,
                                   float* __restrict__ chansum) {
    const int n  = blockIdx.x >> 2;                       // 4 blocks per sample
    const int hw = ((blockIdx.x & 3) << 8) + threadIdx.x; // 0..1023
    const float* px = x + (size_t)n * CCH * HWSZ + hw;
    float s0 = 0.f, s1 = 0.f, s2 = 0.f, s3 = 0.f;
#pragma unroll 4
    for (int c = 0; c < CCH; c += 4) {
        s0 += px[(size_t)(c + 0) * HWSZ];
        s1 += px[(size_t)(c + 1) * HWSZ];
        s2 += px[(size_t)(c + 2) * HWSZ];
        s3 += px[(size_t)(c + 3) * HWSZ];
    }
    chansum[n * HWSZ + hw] = (s0 + s1) + (s2 + s3);
}

// ---------------------------------------------------------------------------
// Kernel 2: per-sample threshold, mask, reduced_x, argmax (first-max),
//           anchor, relative dist/angle. One 1024-thread block per sample.
// ---------------------------------------------------------------------------
__global__ void rcp_persample_kernel(const float* __restrict__ chansum,
                                     float* __restrict__ reduced,   // ws
                                     float* __restrict__ out) {
    __shared__ float sv[HWSZ];
    __shared__ int   si[HWSZ];

    const int n   = blockIdx.x;
    const int tid = threadIdx.x;          // == hw index
    const float v = chansum[n * HWSZ + tid];

    // --- block sum -> threshold ---
    sv[tid] = v;
    __syncthreads();
#pragma unroll
    for (int s = HWSZ / 2; s > 0; s >>= 1) {
        if (tid < s) sv[tid] += sv[tid + s];
        __syncthreads();
    }
    const float thr = sv[0] * (1.0f / (float)HWSZ);
    __syncthreads();

    // --- mask + reduced_x (= bm * chansum / C) ---
    const float bm = (v > thr) ? 1.0f : 0.0f;
    const float r  = bm * v * (1.0f / (float)CCH);
    reduced[n * HWSZ + tid] = r;

    // --- argmax with first-index tiebreak ---
    sv[tid] = r;
    si[tid] = tid;
    __syncthreads();
#pragma unroll
    for (int s = HWSZ / 2; s > 0; s >>= 1) {
        if (tid < s) {
            const float vo = sv[tid + s];
            const int   io = si[tid + s];
            const float vc = sv[tid];
            const int   ic = si[tid];
            if (vo > vc || (vo == vc && io < ic)) { sv[tid] = vo; si[tid] = io; }
        }
        __syncthreads();
    }
    const int idx = si[0];

    // --- anchor + relative dist / angle ---
    const float ai = (float)(idx >> 5);   // label row  = idx / 32
    const float aj = (float)(idx & 31);   // label col  = idx % 32
    const float di = ((float)(tid >> 5) - ai) * (1.0f / SIZEF);
    const float dj = ((float)(tid & 31) - aj) * (1.0f / SIZEF);
    const float dist = sqrtf(di * di + dj * dj) * bm;
    const float ang  = (atan2f(dj, di) * 0.15915494309189535f + 0.5f) * bm;

    out[OUT0_OFF + (size_t)n * (HWSZ * 2) + tid * 2 + 0] = dist;
    out[OUT0_OFF + (size_t)n * (HWSZ * 2) + tid * 2 + 1] = ang;

    if (tid == 0) {
        out[OUT1_OFF + n * 2 + 0] = ai;
        out[OUT1_OFF + n * 2 + 1] = aj;
        out[OUT3_OFF + n]         = (float)idx;   // argmax index as float
    }
}

// ---------------------------------------------------------------------------
// Kernel 3: position_weight[n] = reduced ⊗ reduced via V_WMMA_F32_16X16X4_F32.
// Rank-1 update packed into K=4 WMMA (only K=0 nonzero -> exact f32 result).
// Each wave computes a 16x64 strip (4 tiles, A reused). 65536 waves total.
// A layout (16x4 f32): VGPR0 lanes 0-15 = K=0 column -> rI[m]; rest zero.
// B layout (4x16 f32): VGPR0 lanes 0-15 = K=0 row    -> rJ[n]; rest zero.
// D layout: VGPR p -> row p (lanes 0-15) / row p+8 (lanes 16-31), col = lane%16.
// ---------------------------------------------------------------------------
__global__ void rcp_outer_wmma_kernel(const float* __restrict__ reduced,
                                      float* __restrict__ out) {
    const unsigned wid  = blockIdx.x * 8u + (threadIdx.x >> 5);
    const unsigned lane = threadIdx.x & 31u;

    const unsigned tjg = wid & 15u;         // tile-j group (4 tiles each)
    const unsigned ti  = (wid >> 4) & 63u;  // tile-i
    const unsigned n   = wid >> 10;         // sample

    const float* rn = reduced + n * HWSZ;

    // A operand: rI in lanes 0-15 of first dword (K=0), zeros elsewhere.
    float av = rn[ti * 16 + (lane & 15u)];  // lane&15 keeps loads in-bounds
    av = (lane < 16u) ? av : 0.0f;
    v2f a;
    a.x = av;
    a.y = 0.0f;

    float* outn = out + OUT2_OFF + (size_t)n * (HWSZ * HWSZ);
    const unsigned rbase = ti * 16 + ((lane >> 4) << 3);  // +8 for upper half
    const unsigned cbase = lane & 15u;

#pragma unroll
    for (unsigned t = 0; t < 4; ++t) {
        const unsigned tj = tjg * 4 + t;
        float bv = rn[tj * 16 + (lane & 15u)];
        bv = (lane < 16u) ? bv : 0.0f;
        v2f b;
        b.x = bv;
        b.y = 0.0f;
        v8f c = {};
        // 8 args: (neg_a, A, neg_b, B, c_mod, C, reuse_a, reuse_b)
        v8f d = __builtin_amdgcn_wmma_f32_16x16x4_f32(
            false, a, false, b, (short)0, c, false, false);

        const unsigned col = tj * 16 + cbase;
#pragma unroll
        for (unsigned p = 0; p < 8; ++p) {
            outn[(size_t)(rbase + p) * HWSZ + col] = d[p];
        }
    }
}

// ---------------------------------------------------------------------------
extern "C" void kernel_launch(void* const* d_in, const int* in_sizes, int n_in,
                              void* d_out, int out_size, void* d_ws, size_t ws_size,
                              hipStream_t stream) {
    const float* x = (const float*)d_in[0];
    float* out = (float*)d_out;

    float* chansum = (float*)d_ws;            // N*HW floats   (256 KB)
    float* reduced = chansum + NB * HWSZ;     // N*HW floats   (256 KB)

    // K1: channel sum (reads x once)
    rcp_chansum_kernel<<<NB * (HWSZ / 256), 256, 0, stream>>>(x, chansum);

    // K2: per-sample threshold/mask/argmax/coords
    rcp_persample_kernel<<<NB, HWSZ, 0, stream>>>(chansum, reduced, out);

    // K3: outer product via WMMA: 64 samples * 64 ti * 16 tj-groups = 65536
    // waves; 8 waves (256 threads) per block -> 8192 blocks.
    rcp_outer_wmma_kernel<<<8192, 256, 0, stream>>>(reduced, out);
}